// Attention_28836410425923
// MI455X (gfx1250) — compile-verified
//
#include <hip/hip_runtime.h>
#include <hip/hip_bf16.h>
#include <math.h>

typedef float v2f __attribute__((ext_vector_type(2)));
typedef float v8f __attribute__((ext_vector_type(8)));
typedef int   vi4 __attribute__((vector_size(16)));   // matches builtin's V4i param

#define NUM_HEADS     32
#define NUM_KV_HEADS  8
#define HEAD_DIM      128
#define GQA           4
#define CONTEXT_LEN   8192
#define BLOCK_SZ      16
#define NUM_LBLOCKS   (CONTEXT_LEN / BLOCK_SZ)        // 512 logical blocks
#define KS_STRIDE     132   // 128 + 4 pad: conflict-free A-operand b64 loads
#define VS_STRIDE     136   // 128 + 8 pad: 2*VS_STRIDE % 64 == 16 -> halves hit disjoint banks
#define ATTN_SCALE    0.08838834764831845f
#define PART_STRIDE   (8 + GQA * HEAD_DIM)            // 520 floats per (kvh, part)

#if __has_builtin(__builtin_amdgcn_global_load_async_to_lds_b128)
#define ASYNC_STAGE 1
#if __has_builtin(__builtin_amdgcn_s_wait_asynccnt)
#define WAIT_ASYNC(n) __builtin_amdgcn_s_wait_asynccnt(n)
#else
#define WAIT_ASYNC(n) asm volatile("s_wait_asynccnt %0" ::"n"(n) : "memory")
#endif
#else
#define ASYNC_STAGE 0
#define WAIT_ASYNC(n) __syncthreads()
#endif

// Copy 16 bytes global -> LDS. Async (ASYNCcnt-tracked DMA) when available.
__device__ __forceinline__ void cp16(const float* __restrict__ gsrc, float* ldst) {
#if ASYNC_STAGE
  __builtin_amdgcn_global_load_async_to_lds_b128(
      (__attribute__((address_space(1))) vi4*)gsrc,
      (__attribute__((address_space(3))) vi4*)ldst,
      /*offset=*/0, /*cpol=*/0);
#else
  *(float4*)ldst = *(const float4*)gsrc;
#endif
}

// Stage one 16-token K/V tile (row-major) into LDS; substitute the new token.
__device__ __forceinline__ void stage_tile(
    const float* __restrict__ kc, const float* __restrict__ vc,
    const float* __restrict__ knew, const float* __restrict__ vnew,
    int pb, int tslot, int kvh, int lane,
    float* __restrict__ Ksb, float* __restrict__ Vsb)
{
#pragma unroll 4
  for (int i = 0; i < 16; ++i) {
    const int slot   = pb * BLOCK_SZ + i;
    const size_t row = (size_t)slot * (NUM_KV_HEADS * HEAD_DIM) + (size_t)kvh * HEAD_DIM;
    const float* ks = (slot == tslot) ? (knew + (size_t)kvh * HEAD_DIM) : (kc + row);
    const float* vs = (slot == tslot) ? (vnew + (size_t)kvh * HEAD_DIM) : (vc + row);
    cp16(ks + 4 * lane, Ksb + i * KS_STRIDE + 4 * lane);
    cp16(vs + 4 * lane, Vsb + i * VS_STRIDE + 4 * lane);
  }
}

// ---------------------------------------------------------------------------
// Flash-decode partial kernel: one wave32 per (kv_head, partition).
// Double-buffered async LDS staging; WMMA f32 16x16x4 for QK^T and PV.
// ---------------------------------------------------------------------------
__global__ __launch_bounds__(32) void attn_partial(
    const float* __restrict__ q,    const float* __restrict__ knew,
    const float* __restrict__ vnew, const float* __restrict__ kc,
    const float* __restrict__ vc,   const int* __restrict__ btab,
    const int* __restrict__ smap,   float* __restrict__ ws,
    int num_parts, int tiles_pp)
{
  const int kvh  = blockIdx.x;
  const int part = blockIdx.y;
  const int lane = threadIdx.x;
  const int h    = lane & 15;            // M/N index within a 16-half
  const int off  = (lane < 16) ? 0 : 2;  // K-half offset for A/B layouts
  const bool lo  = lane < 16;

  __shared__ float Ks0[BLOCK_SZ * KS_STRIDE];
  __shared__ float Ks1[BLOCK_SZ * KS_STRIDE];
  __shared__ float Vs0[BLOCK_SZ * VS_STRIDE];
  __shared__ float Vs1[BLOCK_SZ * VS_STRIDE];

  // ---- Preload q in WMMA-B layout: lane h holds q[h][4i+off .. 4i+off+1] ----
  v2f qreg[32];
  if (h < GQA) {
    const float* qp = q + (size_t)(kvh * GQA + h) * HEAD_DIM + off;
#pragma unroll
    for (int i = 0; i < 32; ++i) qreg[i] = *(const v2f*)(qp + 4 * i);
  } else {
#pragma unroll
    for (int i = 0; i < 32; ++i) qreg[i] = (v2f){0.f, 0.f};
  }

  v8f acc[8];
#pragma unroll
  for (int dc = 0; dc < 8; ++dc) acc[dc] = {};
  float m_run = -INFINITY, l_run = 0.f;
  const int tslot = smap[0];
  const int lb0   = part * tiles_pp;

  // Prologue: stage tile 0 into buffer 0.
  stage_tile(kc, vc, knew, vnew, btab[lb0], tslot, kvh, lane, Ks0, Vs0);

  for (int tile = 0; tile < tiles_pp; ++tile) {
    float* Ksb = (tile & 1) ? Ks1 : Ks0;
    float* Vsb = (tile & 1) ? Vs1 : Vs0;
    const bool more = (tile + 1 < tiles_pp);
    if (more) {
      const int pbn = btab[lb0 + tile + 1];
      stage_tile(kc, vc, knew, vnew, pbn, tslot, kvh, lane,
                 (tile & 1) ? Ks0 : Ks1, (tile & 1) ? Vs0 : Vs1);
      // Tile `tile`'s 32 copies complete once only the next tile's 32 remain.
      WAIT_ASYNC(32);
    } else {
      WAIT_ASYNC(0);
    }
    // Prefetch tile+2 (hint; async copy already covers tile+1)
    if (tile + 2 < tiles_pp) {
      const int pb2 = btab[lb0 + tile + 2];
      const size_t r2 = (size_t)(pb2 * BLOCK_SZ + h) * (NUM_KV_HEADS * HEAD_DIM)
                      + (size_t)kvh * HEAD_DIM + (size_t)(lane >> 4) * 64;
      __builtin_prefetch(kc + r2, 0, 0);
      __builtin_prefetch(vc + r2, 0, 0);
    }

    // ---- Stage 1: S[token, head] = K_tile x q^T  (32 x WMMA f32 16x16x4) ----
    v8f s = {};
    const float* abase = &Ksb[h * KS_STRIDE + off];
#pragma unroll
    for (int i = 0; i < 32; ++i) {
      v2f a = *(const v2f*)(abase + 4 * i);
      s = __builtin_amdgcn_wmma_f32_16x16x4_f32(false, a, false, qreg[i],
                                                (short)0, s, false, false);
    }

    // ---- Flash softmax update ----
    float p[8]; float tm = -INFINITY;
#pragma unroll
    for (int r = 0; r < 8; ++r) { p[r] = s[r] * ATTN_SCALE; tm = fmaxf(tm, p[r]); }
    tm = fmaxf(tm, __shfl_xor(tm, 16, 32));
    const float m_new = fmaxf(m_run, tm);
    const float corr  = __expf(m_run - m_new);
    float ts = 0.f;
#pragma unroll
    for (int r = 0; r < 8; ++r) { p[r] = __expf(p[r] - m_new); ts += p[r]; }
    ts += __shfl_xor(ts, 16, 32);
    l_run = l_run * corr + ts;
    m_run = m_new;
    const float c0 = __shfl(corr, 0, 32);
    const float c1 = __shfl(corr, 1, 32);
    const float c2 = __shfl(corr, 2, 32);
    const float c3 = __shfl(corr, 3, 32);
#pragma unroll
    for (int dc = 0; dc < 8; ++dc) {
      acc[dc][0] *= c0; acc[dc][1] *= c1; acc[dc][2] *= c2; acc[dc][3] *= c3;
    }

    // ---- Build A = P^T in WMMA-A layout via half-swap shuffles ----
    float pq[8];
#pragma unroll
    for (int r = 0; r < 8; ++r) pq[r] = __shfl_xor(p[r], 16, 32);
    v2f A[4];
    A[0].x = lo ? p[0]  : pq[2];  A[0].y = lo ? p[1]  : pq[3];
    A[1].x = lo ? p[4]  : pq[6];  A[1].y = lo ? p[5]  : pq[7];
    A[2].x = lo ? pq[0] : p[2];   A[2].y = lo ? pq[1] : p[3];
    A[3].x = lo ? pq[4] : p[6];   A[3].y = lo ? pq[5] : p[7];

    // ---- Stage 2: O[head, dim] += P^T x V  (32 x WMMA f32 16x16x4) ----
    // V is row-major: B[K=token 4tc+off+{0,1}, N=dim dc*16+h]
    const float* vbase = &Vsb[off * VS_STRIDE + h];
#pragma unroll
    for (int tc = 0; tc < 4; ++tc) {
#pragma unroll
      for (int dc = 0; dc < 8; ++dc) {
        v2f b;
        b.x = vbase[tc * 4 * VS_STRIDE + dc * 16];
        b.y = vbase[tc * 4 * VS_STRIDE + VS_STRIDE + dc * 16];
        acc[dc] = __builtin_amdgcn_wmma_f32_16x16x4_f32(false, A[tc], false, b,
                                                        (short)0, acc[dc], false, false);
      }
    }
  }

  // ---- Emit partials: m[4], l[4], out[4][128] ----
  float* base = ws + (size_t)(kvh * num_parts + part) * PART_STRIDE;
  if (lane < GQA) { base[lane] = m_run; base[GQA + lane] = l_run; }
  if (lo) {
#pragma unroll
    for (int dc = 0; dc < 8; ++dc)
#pragma unroll
      for (int r = 0; r < GQA; ++r)
        base[8 + r * HEAD_DIM + dc * 16 + lane] = acc[dc][r];
  }
}

// ---------------------------------------------------------------------------
// Combine partitions with softmax rescaling, then normalize.
// ---------------------------------------------------------------------------
__global__ __launch_bounds__(128) void attn_reduce(const float* __restrict__ ws,
                                                   float* __restrict__ out,
                                                   int num_parts)
{
  const int hq  = blockIdx.x;
  const int kvh = hq >> 2, g = hq & 3;
  const int d   = threadIdx.x;
  const float* b0 = ws + (size_t)kvh * num_parts * PART_STRIDE;
  float mg = -INFINITY;
  for (int p = 0; p < num_parts; ++p) mg = fmaxf(mg, b0[p * PART_STRIDE + g]);
  float num = 0.f, den = 0.f;
  for (int p = 0; p < num_parts; ++p) {
    const float* bp = b0 + p * PART_STRIDE;
    const float w = __expf(bp[g] - mg);
    den += w * bp[GQA + g];
    num += w * bp[8 + g * HEAD_DIM + d];
  }
  out[hq * HEAD_DIM + d] = num / den;
}

extern "C" void kernel_launch(void* const* d_in, const int* in_sizes, int n_in,
                              void* d_out, int out_size, void* d_ws, size_t ws_size,
                              hipStream_t stream) {
  const float* q    = (const float*)d_in[0];
  const float* knew = (const float*)d_in[1];
  const float* vnew = (const float*)d_in[2];
  const float* kc   = (const float*)d_in[3];
  const float* vc   = (const float*)d_in[4];
  const int*   btab = (const int*)d_in[5];
  const int*   smap = (const int*)d_in[6];
  float* ws = (float*)d_ws;

  // Pick the most parallel partitioning the workspace allows (deterministic).
  int parts = 64;                                       // 512 waves when possible
  while (parts > 1 &&
         ws_size < (size_t)NUM_KV_HEADS * parts * PART_STRIDE * sizeof(float))
    parts >>= 1;
  const int tiles_pp = NUM_LBLOCKS / parts;

  attn_partial<<<dim3(NUM_KV_HEADS, parts), 32, 0, stream>>>(
      q, knew, vnew, kc, vc, btab, smap, ws, parts, tiles_pp);
  attn_reduce<<<NUM_HEADS, HEAD_DIM, 0, stream>>>(ws, (float*)d_out, parts);
}